// GINEncoder_41455024341694
// MI455X (gfx1250) — compile-verified
//
#include <hip/hip_runtime.h>

#define HDIM 128
#define FDIM 9
#define BN_EPS 1e-5f

typedef __attribute__((ext_vector_type(2))) float v2f;
typedef __attribute__((ext_vector_type(8))) float v8f;

// ---------------- copies / passthrough ----------------
__global__ __launch_bounds__(256) void copy_f4_k(const float4* __restrict__ s,
                                                 float4* __restrict__ d, int n4) {
  int i = blockIdx.x * 256 + threadIdx.x;
  if (i < n4) d[i] = s[i];
}

__global__ __launch_bounds__(256) void copy_f_k(const float* __restrict__ s,
                                                float* __restrict__ d, int n) {
  int i = blockIdx.x * 256 + threadIdx.x;
  if (i < n) d[i] = s[i];
}

__global__ __launch_bounds__(256) void batch_out_k(const int* __restrict__ b,
                                                   float* __restrict__ o, int n) {
  int i = blockIdx.x * 256 + threadIdx.x;
  if (i < n) o[i] = (float)b[i];
}

// ---------------- GIN edge aggregation: z[dst] += h[src] ----------------
// One wave per edge; each lane moves a float4 (32 lanes * 4 = 128 channels).
// h is L2-resident (51.2 MB << 192 MB), atomics resolve at L2.
__global__ __launch_bounds__(256) void scatter_h128_k(const float* __restrict__ h,
                                                      const int* __restrict__ src,
                                                      const int* __restrict__ dst,
                                                      float* __restrict__ z, int E) {
  int gt = blockIdx.x * 256 + threadIdx.x;
  int e = gt >> 5, lane = gt & 31;
  if (e >= E) return;
  int s = src[e], d = dst[e];
  float4 v = ((const float4*)(h + (size_t)s * HDIM))[lane];
  float* zp = z + (size_t)d * HDIM + lane * 4;
  atomicAdd(zp + 0, v.x);
  atomicAdd(zp + 1, v.y);
  atomicAdd(zp + 2, v.z);
  atomicAdd(zp + 3, v.w);
}

// Layer-0 aggregation over the 9 raw features.
__global__ __launch_bounds__(256) void scatter_f9_k(const float* __restrict__ x,
                                                    const int* __restrict__ src,
                                                    const int* __restrict__ dst,
                                                    float* __restrict__ z0, int E) {
  int gt = blockIdx.x * 256 + threadIdx.x;
  int e = gt >> 4, c = gt & 15;
  if (e >= E || c >= FDIM) return;
  atomicAdd(&z0[(size_t)dst[e] * FDIM + c], x[(size_t)src[e] * FDIM + c]);
}

// ---------------- layer-0 MLP1 (9 -> 128): K too small for WMMA, plain VALU ----
__global__ __launch_bounds__(256) void mlp1_f9_k(const float* __restrict__ z0,
                                                 const float* __restrict__ w1,
                                                 const float* __restrict__ b1,
                                                 float* __restrict__ y1, int n) {
  int t = blockIdx.x * 256 + threadIdx.x;
  if (t >= n * HDIM) return;
  int i = t >> 7, o = t & (HDIM - 1);
  const float* zr = z0 + (size_t)i * FDIM;
  float acc = b1[o];
#pragma unroll
  for (int k = 0; k < FDIM; ++k) acc = fmaf(zr[k], w1[k * HDIM + o], acc);
  y1[t] = fmaxf(acc, 0.0f);
}

// ---------------- Y = [relu](A(n x 128) @ W(128 x 128) + bias) -----------------
// fp32 WMMA 16x16x4. Block = 8 waves = 64 rows x 128 cols; A tile staged in LDS.
// Per wave: 16 cols, 4 row-subtiles, 32 k-steps -> 128 v_wmma_f32_16x16x4_f32.
__global__ __launch_bounds__(256) void gemm_h128_k(const float* __restrict__ A,
                                                   const float* __restrict__ W,
                                                   const float* __restrict__ bias,
                                                   float* __restrict__ Y,
                                                   int n, int relu) {
  __shared__ float As[64][HDIM + 4];
  const int tid = threadIdx.x;
  const int row0 = blockIdx.x * 64;
  for (int idx = tid; idx < 64 * HDIM; idx += 256) {
    int r = idx >> 7, c = idx & (HDIM - 1);
    int g = row0 + r;
    As[r][c] = (g < n) ? A[(size_t)g * HDIM + c] : 0.0f;
  }
  __syncthreads();

  const int lane = tid & 31;
  const int wave = tid >> 5;
  const int col0 = wave * 16;
  const int half = lane >> 4;   // lanes 16-31 hold k+2,k+3
  const int mn = lane & 15;     // M index for A, N index for B/C/D

  v8f acc[4] = {};
  for (int kk = 0; kk < HDIM; kk += 4) {
    const int k2 = kk + half * 2;
    v2f b;                                    // B 4x16: b.x=B[k2][n], b.y=B[k2+1][n]
    b.x = W[(size_t)k2 * HDIM + col0 + mn];
    b.y = W[(size_t)(k2 + 1) * HDIM + col0 + mn];
#pragma unroll
    for (int rt = 0; rt < 4; ++rt) {          // A 16x4: a.x=A[m][k2], a.y=A[m][k2+1]
      v2f a = *(const v2f*)&As[rt * 16 + mn][k2];
      acc[rt] = __builtin_amdgcn_wmma_f32_16x16x4_f32(
          false, a, false, b, (short)0, acc[rt], false, false);
    }
  }

  const float bv = bias[col0 + mn];
#pragma unroll
  for (int rt = 0; rt < 4; ++rt) {
#pragma unroll
    for (int j = 0; j < 8; ++j) {             // C/D: VGPR j -> row j + 8*half
      int m = j + half * 8;
      int g = row0 + rt * 16 + m;
      if (g < n) {
        float v = acc[rt][j] + bv;
        if (relu) v = fmaxf(v, 0.0f);
        Y[(size_t)g * HDIM + col0 + mn] = v;
      }
    }
  }
}

// ---------------- BatchNorm (training-mode batch stats) ----------------
__global__ void zero_stats_k(float* sums, float* sqs) {
  int i = threadIdx.x;
  if (i < HDIM) { sums[i] = 0.0f; sqs[i] = 0.0f; }
}

__global__ __launch_bounds__(256) void bn_stats_k(const float* __restrict__ t,
                                                  float* __restrict__ sums,
                                                  float* __restrict__ sqs, int n) {
  int c = threadIdx.x & (HDIM - 1);
  int r = blockIdx.x * 2 + (threadIdx.x >> 7);
  int stride = gridDim.x * 2;
  float s = 0.0f, q = 0.0f;
  for (; r < n; r += stride) {
    float v = t[(size_t)r * HDIM + c];
    s += v;
    q += v * v;
  }
  atomicAdd(&sums[c], s);
  atomicAdd(&sqs[c], q);
}

__global__ void bn_final_k(const float* sums, const float* sqs,
                           const float* gamma, const float* beta,
                           float* sc, float* sh, int n) {
  int c = threadIdx.x;
  if (c >= HDIM) return;
  float inv_n = 1.0f / (float)n;
  float mean = sums[c] * inv_n;
  float var = sqs[c] * inv_n - mean * mean;   // biased variance (jnp.var default)
  float s = gamma[c] * rsqrtf(var + BN_EPS);
  sc[c] = s;
  sh[c] = beta[c] - mean * s;
}

__global__ __launch_bounds__(256) void bn_apply_k(const float4* __restrict__ t,
                                                  const float* __restrict__ sc,
                                                  const float* __restrict__ sh,
                                                  float4* __restrict__ h, int n4) {
  int i = blockIdx.x * 256 + threadIdx.x;
  if (i >= n4) return;
  int c0 = (i * 4) & (HDIM - 1);
  float4 v = t[i];
  float4 s = *(const float4*)&sc[c0];
  float4 b = *(const float4*)&sh[c0];
  float4 o;
  o.x = fmaxf(fmaf(v.x, s.x, b.x), 0.0f);
  o.y = fmaxf(fmaf(v.y, s.y, b.y), 0.0f);
  o.z = fmaxf(fmaf(v.z, s.z, b.z), 0.0f);
  o.w = fmaxf(fmaf(v.w, s.w, b.w), 0.0f);
  h[i] = o;
}

// ------------------------------------------------------------------------------
static inline int cdiv(long long a, long long b) { return (int)((a + b - 1) / b); }

extern "C" void kernel_launch(void* const* d_in, const int* in_sizes, int n_in,
                              void* d_out, int out_size, void* d_ws, size_t ws_size,
                              hipStream_t stream) {
  (void)n_in; (void)out_size; (void)ws_size;

  const float* x       = (const float*)d_in[0];
  const int*   ei      = (const int*)d_in[1];
  const int*   batch   = (const int*)d_in[2];
  const float* w1_0    = (const float*)d_in[3];
  const float* b1_0    = (const float*)d_in[4];
  const float* w2_0    = (const float*)d_in[5];
  const float* b2_0    = (const float*)d_in[6];
  const float* gamma_0 = (const float*)d_in[7];
  const float* beta_0  = (const float*)d_in[8];
  const float* ws1     = (const float*)d_in[9];
  const float* bs1     = (const float*)d_in[10];
  const float* ws2     = (const float*)d_in[11];
  const float* bs2     = (const float*)d_in[12];
  const float* gammas  = (const float*)d_in[13];
  const float* betas   = (const float*)d_in[14];

  const int N = in_sizes[0] / FDIM;
  const int E = in_sizes[1] / 2;
  const int* srcI = ei;        // edge_index[0]
  const int* dstI = ei + E;    // edge_index[1]

  // h lives in d_out (first N*H floats); batch passthrough in the tail.
  float* h    = (float*)d_out;
  float* z    = (float*)d_ws;                       // N*H
  float* y1   = z + (size_t)N * HDIM;               // N*H
  float* z0   = y1 + (size_t)N * HDIM;              // N*F (layer 0 only)
  float* sums = z0 + (size_t)N * FDIM;              // 128
  float* sqs  = sums + HDIM;                        // 128
  float* sc   = sqs + HDIM;                         // 128
  float* sh   = sc + HDIM;                          // 128
  // workspace use: (2*N*H + N*F + 512)*4 B ~= 106 MB

  const int nh4 = (N * HDIM) / 4;

  batch_out_k<<<cdiv(N, 256), 256, 0, stream>>>(batch, h + (size_t)N * HDIM, N);

  // ---------------- layer 0 (F=9 -> H -> H) ----------------
  copy_f_k<<<cdiv((long long)N * FDIM, 256), 256, 0, stream>>>(x, z0, N * FDIM);
  scatter_f9_k<<<cdiv((long long)E * 16, 256), 256, 0, stream>>>(x, srcI, dstI, z0, E);
  mlp1_f9_k<<<cdiv((long long)N * HDIM, 256), 256, 0, stream>>>(z0, w1_0, b1_0, y1, N);
  gemm_h128_k<<<cdiv(N, 64), 256, 0, stream>>>(y1, w2_0, b2_0, z, N, 0);
  zero_stats_k<<<1, 256, 0, stream>>>(sums, sqs);
  bn_stats_k<<<512, 256, 0, stream>>>(z, sums, sqs, N);
  bn_final_k<<<1, 128, 0, stream>>>(sums, sqs, gamma_0, beta_0, sc, sh, N);
  bn_apply_k<<<cdiv(nh4, 256), 256, 0, stream>>>((const float4*)z, sc, sh, (float4*)h, nh4);

  // ---------------- layers 1..4 (H -> H -> H) ----------------
  for (int l = 0; l < 4; ++l) {
    const float* w1 = ws1 + (size_t)l * HDIM * HDIM;
    const float* b1 = bs1 + (size_t)l * HDIM;
    const float* w2 = ws2 + (size_t)l * HDIM * HDIM;
    const float* b2 = bs2 + (size_t)l * HDIM;
    const float* ga = gammas + (size_t)l * HDIM;
    const float* be = betas + (size_t)l * HDIM;

    copy_f4_k<<<cdiv(nh4, 256), 256, 0, stream>>>((const float4*)h, (float4*)z, nh4);
    scatter_h128_k<<<cdiv((long long)E * 32, 256), 256, 0, stream>>>(h, srcI, dstI, z, E);
    gemm_h128_k<<<cdiv(N, 64), 256, 0, stream>>>(z, w1, b1, y1, N, 1);
    gemm_h128_k<<<cdiv(N, 64), 256, 0, stream>>>(y1, w2, b2, z, N, 0);
    zero_stats_k<<<1, 256, 0, stream>>>(sums, sqs);
    bn_stats_k<<<512, 256, 0, stream>>>(z, sums, sqs, N);
    bn_final_k<<<1, 128, 0, stream>>>(sums, sqs, ga, be, sc, sh, N);
    bn_apply_k<<<cdiv(nh4, 256), 256, 0, stream>>>((const float4*)z, sc, sh, (float4*)h, nh4);
  }
}